// PaGCN_ogb_54065048323073
// MI455X (gfx1250) — compile-verified
//
#include <hip/hip_runtime.h>
#include <math.h>

typedef __attribute__((ext_vector_type(2))) float v2f;
typedef __attribute__((ext_vector_type(8))) float v8f;

#define NF 128   // feature width for hidden layers (NFEAT == NHID == 128)

// ---------------------------------------------------------------------------
// zero-fill (d_ws is not re-poisoned between replays -> must clear every call)
// ---------------------------------------------------------------------------
__global__ void fill_zero(float* __restrict__ p, long n) {
  long i = (long)blockIdx.x * blockDim.x + threadIdx.x;
  long stride = (long)gridDim.x * blockDim.x;
  for (; i < n; i += stride) p[i] = 0.0f;
}

// ---------------------------------------------------------------------------
// SPMM: out[dst] += w * in[src], w = vals[e] * (M[src]?) * (AM[dst]?)
// one 128-thread block per edge; edge scalars are block-uniform -> s_loads;
// 32 lanes/wave hit 32 consecutive floats -> coalesced 128B atomic bursts.
// ---------------------------------------------------------------------------
__global__ __launch_bounds__(128) void spmm_kernel(
    const float* __restrict__ X, const int* __restrict__ src,
    const int* __restrict__ dst, const float* __restrict__ vals,
    const float* __restrict__ Mv, const float* __restrict__ AMv,
    float* __restrict__ out) {
  int e = blockIdx.x;
  int f = threadIdx.x;
  int s = src[e];
  int d = dst[e];
  float w = vals[e];
  if (Mv)  w *= Mv[s];   // layer-0: fold M (per-source) into edge weight
  if (AMv) w *= AMv[d];  // layer-0: fold AM (per-dest) into edge weight
  atomicAdd(out + (size_t)d * NF + f, w * X[(size_t)s * NF + f]);
}

// ---------------------------------------------------------------------------
// out[r, c] = relu?( bn?( sum_k A[r,k]*W[c,k] + bias[c] ) )
// One wave per 16x16 tile; K=128 as 32 x v_wmma_f32_16x16x4_f32.
// A layout (16x4 f32): lanes 0-15 -> {K=k0,k0+1}, lanes 16-31 -> {K=k0+2,k0+3}
// B = W^T uses the mirrored layout; C/D: VGPR i = row (i | lane[4]<<3), col = lane&15.
// ---------------------------------------------------------------------------
__global__ __launch_bounds__(256) void gemm_wmma_kernel(
    const float* __restrict__ A,      // [n_rows x 128]
    const float* __restrict__ W,      // [n_out_real x 128]
    const float* __restrict__ bias,   // [n_out_real]
    const float* __restrict__ g, const float* __restrict__ be,
    const float* __restrict__ rm, const float* __restrict__ rv,
    float* __restrict__ out,          // [n_rows x n_out_real]
    int n_rows, int n_out_real, int tiles_n, int relu) {
  const int K = NF;
  int lane = threadIdx.x & 31;
  int wave = threadIdx.x >> 5;
  int tiles_m = (n_rows + 15) >> 4;
  int tile = blockIdx.x * 8 + wave;
  if (tile >= tiles_m * tiles_n) return;  // wave-uniform: EXEC stays all-ones
  int tm = tile / tiles_n;
  int tn = tile - tm * tiles_n;

  int hl = lane >> 4;   // half-wave select (0/1)
  int ll = lane & 15;
  int row = tm * 16 + ll;
  int ar  = row < n_rows ? row : n_rows - 1;      // clamp (N=100000 is /16 anyway)
  int col = tn * 16 + ll;
  int cw  = col < n_out_real ? col : n_out_real - 1;  // clamp padded cols (40->48)
  int kh  = hl * 2;

  const float* ap = A + (size_t)ar * K + kh;  // 8B aligned (kh even, k0 mult of 4)
  const float* wp = W + (size_t)cw * K + kh;

  v8f c = {};
#pragma unroll
  for (int k0 = 0; k0 < K; k0 += 4) {
    v2f a = *(const v2f*)(ap + k0);
    v2f b = *(const v2f*)(wp + k0);
    c = __builtin_amdgcn_wmma_f32_16x16x4_f32(
        /*neg_a=*/false, a, /*neg_b=*/false, b,
        /*c_mod=*/(short)0, c, /*reuse_a=*/false, /*reuse_b=*/false);
  }

  // per-lane column constants (all 8 C VGPRs share one column per lane)
  float bb = bias ? bias[cw] : 0.0f;
  float sc = 1.0f, sh = 0.0f;
  if (g) {
    float s = g[cw] * rsqrtf(rv[cw] + 1e-5f);
    sc = s;
    sh = be[cw] - rm[cw] * s;
  }
  if (col < n_out_real) {
    int rb = tm * 16 + hl * 8;
#pragma unroll
    for (int i = 0; i < 8; ++i) {
      int r = rb + i;
      if (r < n_rows) {
        float v = (c[i] + bb) * sc + sh;
        if (relu) v = fmaxf(v, 0.0f);
        out[(size_t)r * n_out_real + col] = v;
      }
    }
  }
}

// ---------------------------------------------------------------------------
// in-place log_softmax over 40 classes, one thread per node
// ---------------------------------------------------------------------------
__global__ __launch_bounds__(256) void log_softmax40(float* __restrict__ out, int n) {
  int i = blockIdx.x * blockDim.x + threadIdx.x;
  if (i >= n) return;
  float* row = out + (size_t)i * 40;
  float m = row[0];
#pragma unroll
  for (int j = 1; j < 40; ++j) m = fmaxf(m, row[j]);
  float s = 0.0f;
#pragma unroll
  for (int j = 0; j < 40; ++j) s += expf(row[j] - m);
  float l = m + logf(s);
#pragma unroll
  for (int j = 0; j < 40; ++j) row[j] = row[j] - l;
}

extern "C" void kernel_launch(void* const* d_in, const int* in_sizes, int n_in,
                              void* d_out, int out_size, void* d_ws, size_t ws_size,
                              hipStream_t stream) {
  const float* x     = (const float*)d_in[0];
  const float* Mv    = (const float*)d_in[1];
  const float* AMv   = (const float*)d_in[2];
  const int*   srcZ  = (const int*)d_in[3];
  const int*   dstZ  = (const int*)d_in[4];
  const float* valsZ = (const float*)d_in[5];
  const int*   src   = (const int*)d_in[6];
  const int*   dst   = (const int*)d_in[7];
  const float* vals  = (const float*)d_in[8];
  const float* W0 = (const float*)d_in[9];
  const float* b0 = (const float*)d_in[10];
  const float* g0 = (const float*)d_in[11];
  const float* be0 = (const float*)d_in[12];
  const float* rm0 = (const float*)d_in[13];
  const float* rv0 = (const float*)d_in[14];
  const float* W1 = (const float*)d_in[15];
  const float* b1 = (const float*)d_in[16];
  const float* g1 = (const float*)d_in[17];
  const float* be1 = (const float*)d_in[18];
  const float* rm1 = (const float*)d_in[19];
  const float* rv1 = (const float*)d_in[20];
  const float* W2 = (const float*)d_in[21];
  const float* b2 = (const float*)d_in[22];

  int N = in_sizes[1];  // M is (N,1)
  int E = in_sizes[3];  // srcZ is (E,)
  float* bufA = (float*)d_ws;                  // aggregation buffer [N x 128]
  float* bufB = bufA + (size_t)N * NF;         // hidden buffer      [N x 128]
  float* outp = (float*)d_out;                 // logits / log-probs [N x 40]

  int tiles_m = (N + 15) / 16;
  long nfeat_elems = (long)N * NF;
  int gemm_blocks_h = (tiles_m * 8 + 7) / 8;   // 128 out cols -> 8 col-tiles
  int gemm_blocks_o = (tiles_m * 3 + 7) / 8;   // 40 out cols -> 3 col-tiles (padded)

  // ---- layer 0: PaGConv ----
  fill_zero<<<2048, 256, 0, stream>>>(bufA, nfeat_elems);
  spmm_kernel<<<E, 128, 0, stream>>>(x, srcZ, dstZ, valsZ, Mv, AMv, bufA);
  gemm_wmma_kernel<<<gemm_blocks_h, 256, 0, stream>>>(
      bufA, W0, b0, g0, be0, rm0, rv0, bufB, N, NF, 8, 1);

  // ---- layer 1: GConv ----
  fill_zero<<<2048, 256, 0, stream>>>(bufA, nfeat_elems);
  spmm_kernel<<<E, 128, 0, stream>>>(bufB, src, dst, vals, nullptr, nullptr, bufA);
  gemm_wmma_kernel<<<gemm_blocks_h, 256, 0, stream>>>(
      bufA, W1, b1, g1, be1, rm1, rv1, bufB, N, NF, 8, 1);

  // ---- layer 2: GConv -> logits -> log_softmax ----
  fill_zero<<<2048, 256, 0, stream>>>(bufA, nfeat_elems);
  spmm_kernel<<<E, 128, 0, stream>>>(bufB, src, dst, vals, nullptr, nullptr, bufA);
  gemm_wmma_kernel<<<gemm_blocks_o, 256, 0, stream>>>(
      bufA, W2, b2, nullptr, nullptr, nullptr, nullptr, outp, N, 40, 3, 0);
  log_softmax40<<<(N + 255) / 256, 256, 0, stream>>>(outp, N);
}